// WholeGNN_40896678592679
// MI455X (gfx1250) — compile-verified
//
#include <hip/hip_runtime.h>
#include <math.h>

typedef __attribute__((ext_vector_type(2))) float v2f;
typedef __attribute__((ext_vector_type(8))) float v8f;

#define D 64

// ---------------- degree / norm prep ----------------

__global__ void k_init_deg(float* __restrict__ deg, int N) {
  int i = blockIdx.x * blockDim.x + threadIdx.x;
  if (i < N) deg[i] = 1.0f;  // self-loop weight 1
}

__global__ void k_acc_deg(const int* __restrict__ col, const float* __restrict__ ew,
                          float* __restrict__ deg, int E) {
  int e = blockIdx.x * blockDim.x + threadIdx.x;
  if (e < E) atomicAdd(&deg[col[e]], ew[e]);
}

__global__ void k_dinv(float* __restrict__ deg, int N) {
  int i = blockIdx.x * blockDim.x + threadIdx.x;
  if (i < N) {
    float d = deg[i];
    deg[i] = (d > 0.0f) ? rsqrtf(d) : 0.0f;  // in-place: deg -> dinv
  }
}

__global__ void k_norm(const int* __restrict__ row, const int* __restrict__ col,
                       const float* __restrict__ ew, const float* __restrict__ dinv,
                       float* __restrict__ norm, int E) {
  int e = blockIdx.x * blockDim.x + threadIdx.x;
  if (e < E) norm[e] = dinv[row[e]] * ew[e] * dinv[col[e]];
}

// ---------------- fp32 WMMA GEMM:  C[M,64] = A[M,64] @ W[64,64] (+bias) ----------------
// One wave computes a full 16x64 output stripe. W staged in LDS per block.
// M must be a multiple of 16 (50000 = 3125*16). EXEC is all-ones per active wave.

__global__ __launch_bounds__(256) void k_gemm64(const float* __restrict__ A,
                                                const float* __restrict__ W,
                                                const float* __restrict__ bias,
                                                float* __restrict__ C, int M) {
  __shared__ float Ws[D * D];
  for (int t = threadIdx.x; t < D * D / 4; t += blockDim.x)
    ((float4*)Ws)[t] = ((const float4*)W)[t];
  __syncthreads();

  int lane  = threadIdx.x & 31;
  int mtile = blockIdx.x * (blockDim.x >> 5) + (threadIdx.x >> 5);
  if (mtile * 16 >= M) return;  // wave-uniform: EXEC stays all-ones for WMMA

  int m  = lane & 15;   // A row within tile / B,N column within tile
  int kg = lane >> 4;   // K-group (0: K=0,1  1: K=2,3)

  const float* Arow = A + (size_t)(mtile * 16 + m) * D;
  v8f acc0 = {}, acc1 = {}, acc2 = {}, acc3 = {};

#pragma unroll
  for (int k0 = 0; k0 < D; k0 += 4) {
    int ka = k0 + kg * 2;
    v2f a;
    a.x = Arow[ka];
    a.y = Arow[ka + 1];
    const float* Wr0 = Ws + ka * D + m;        // B[k][n], n = m lane pos
    const float* Wr1 = Ws + (ka + 1) * D + m;
    v2f b0, b1, b2, b3;
    b0.x = Wr0[0];  b0.y = Wr1[0];
    b1.x = Wr0[16]; b1.y = Wr1[16];
    b2.x = Wr0[32]; b2.y = Wr1[32];
    b3.x = Wr0[48]; b3.y = Wr1[48];
    acc0 = __builtin_amdgcn_wmma_f32_16x16x4_f32(false, a, false, b0, (short)0, acc0, false, false);
    acc1 = __builtin_amdgcn_wmma_f32_16x16x4_f32(false, a, false, b1, (short)0, acc1, false, false);
    acc2 = __builtin_amdgcn_wmma_f32_16x16x4_f32(false, a, false, b2, (short)0, acc2, false, false);
    acc3 = __builtin_amdgcn_wmma_f32_16x16x4_f32(false, a, false, b3, (short)0, acc3, false, false);
  }

  // D layout: lane L, vgpr v -> row = v + 8*(L>>4), col = ntile*16 + (L&15)
  int rbase = mtile * 16 + 8 * (lane >> 4);
  int n     = lane & 15;
  v8f accs[4] = {acc0, acc1, acc2, acc3};
#pragma unroll
  for (int nt = 0; nt < 4; ++nt) {
    int c = nt * 16 + n;
    float badd = bias ? bias[c] : 0.0f;
#pragma unroll
    for (int v = 0; v < 8; ++v)
      C[(size_t)(rbase + v) * D + c] = accs[nt][v] + badd;
  }
}

// ---------------- aggregation ----------------

// out[i][d] = dinv[i]^2 * hw[i][d] + b[d]   (self-loop term + bias)
__global__ void k_agg_init(const float* __restrict__ hw, const float* __restrict__ dinv,
                           const float* __restrict__ b, float* __restrict__ out, int N) {
  int t = blockIdx.x * blockDim.x + threadIdx.x;
  if (t < N * D) {
    int i = t >> 6, d = t & 63;
    float di = dinv[i];
    out[t] = di * di * hw[t] + b[d];
  }
}

// one wave per edge, float2 per lane: out[col] += norm[e] * hw[row]
__global__ void k_agg_edges(const int* __restrict__ row, const int* __restrict__ col,
                            const float* __restrict__ norm, const float* __restrict__ hw,
                            float* __restrict__ out, int E) {
  int t = blockIdx.x * blockDim.x + threadIdx.x;
  int e = t >> 5, lane = t & 31;
  if (e >= E) return;
  int r = row[e], c = col[e];
  float w = norm[e];
  float2 v = ((const float2*)(hw + (size_t)r * D))[lane];
  float* op = out + (size_t)c * D + lane * 2;
  atomicAdd(op, w * v.x);
  atomicAdd(op + 1, w * v.y);
}

__global__ void k_relu(float* __restrict__ x, int n) {
  int t = blockIdx.x * blockDim.x + threadIdx.x;
  if (t < n) x[t] = fmaxf(x[t], 0.0f);
}

// ---------------- fusion head: sigmoid([h | f] @ Wo + bo) ----------------

__global__ void k_head(const float* __restrict__ h, const float* __restrict__ f,
                       const float* __restrict__ Wo, const float* __restrict__ bo,
                       float* __restrict__ out, int N) {
  int i = blockIdx.x * blockDim.x + threadIdx.x;
  if (i >= N) return;
  const float4* h4 = (const float4*)(h + (size_t)i * D);
  const float4* f4 = (const float4*)(f + (size_t)i * D);
  const float4* w4 = (const float4*)Wo;
  float s = bo[0];
#pragma unroll
  for (int k = 0; k < 16; ++k) {
    float4 a = h4[k], w = w4[k];
    s = fmaf(a.x, w.x, fmaf(a.y, w.y, fmaf(a.z, w.z, fmaf(a.w, w.w, s))));
  }
#pragma unroll
  for (int k = 0; k < 16; ++k) {
    float4 a = f4[k], w = w4[16 + k];
    s = fmaf(a.x, w.x, fmaf(a.y, w.y, fmaf(a.z, w.z, fmaf(a.w, w.w, s))));
  }
  out[i] = 1.0f / (1.0f + __expf(-s));
}

// ---------------- launch ----------------

extern "C" void kernel_launch(void* const* d_in, const int* in_sizes, int n_in,
                              void* d_out, int out_size, void* d_ws, size_t ws_size,
                              hipStream_t stream) {
  const float* x    = (const float*)d_in[0];
  const float* flat = (const float*)d_in[1];
  const int*   eidx = (const int*)d_in[2];
  const float* ew   = (const float*)d_in[3];
  const float* W1   = (const float*)d_in[4];
  const float* b1   = (const float*)d_in[5];
  const float* W2   = (const float*)d_in[6];
  const float* b2   = (const float*)d_in[7];
  const float* Wf   = (const float*)d_in[8];
  const float* bf   = (const float*)d_in[9];
  const float* Wo   = (const float*)d_in[10];
  const float* bo   = (const float*)d_in[11];
  float* out = (float*)d_out;

  int N = in_sizes[0] / D;
  int E = in_sizes[3];
  const int* row = eidx;       // edge_index[0]
  const int* col = eidx + E;   // edge_index[1]

  float* ws   = (float*)d_ws;
  float* dinv = ws;                       ws += N;
  float* norm = ws;                       ws += E;
  float* bufA = ws;                       ws += (size_t)N * D;  // hw scratch
  float* bufH = ws;                       ws += (size_t)N * D;  // conv out
  float* bufF = ws;                       ws += (size_t)N * D;  // flat branch

  dim3 blk(256);
  auto cdiv = [](long long a, long long b) { return (unsigned)((a + b - 1) / b); };
  long long ND = (long long)N * D;
  unsigned gemmBlocks = cdiv(N / 16, 8);  // 8 waves/block, 1 mtile/wave

  // norm prep
  k_init_deg<<<cdiv(N, 256), blk, 0, stream>>>(dinv, N);
  k_acc_deg<<<cdiv(E, 256), blk, 0, stream>>>(col, ew, dinv, E);
  k_dinv<<<cdiv(N, 256), blk, 0, stream>>>(dinv, N);
  k_norm<<<cdiv(E, 256), blk, 0, stream>>>(row, col, ew, dinv, norm, E);

  // layer 1: relu(gcn_conv(x, W1, b1))
  k_gemm64<<<gemmBlocks, blk, 0, stream>>>(x, W1, nullptr, bufA, N);
  k_agg_init<<<cdiv(ND, 256), blk, 0, stream>>>(bufA, dinv, b1, bufH, N);
  k_agg_edges<<<cdiv((long long)E * 32, 256), blk, 0, stream>>>(row, col, norm, bufA, bufH, E);
  k_relu<<<cdiv(ND, 256), blk, 0, stream>>>(bufH, (int)ND);

  // layer 2: gcn_conv(h, W2, b2)
  k_gemm64<<<gemmBlocks, blk, 0, stream>>>(bufH, W2, nullptr, bufA, N);
  k_agg_init<<<cdiv(ND, 256), blk, 0, stream>>>(bufA, dinv, b2, bufH, N);
  k_agg_edges<<<cdiv((long long)E * 32, 256), blk, 0, stream>>>(row, col, norm, bufA, bufH, E);

  // flat branch: flat @ Wf + bf
  k_gemm64<<<gemmBlocks, blk, 0, stream>>>(flat, Wf, bf, bufF, N);

  // head
  k_head<<<cdiv(N, 256), blk, 0, stream>>>(bufH, bufF, Wo, bo, out, N);
}